// MaskCostVolume_42219528520113
// MI455X (gfx1250) — compile-verified
//
#include <hip/hip_runtime.h>
#include <math.h>

typedef __attribute__((ext_vector_type(2))) float v2f;
typedef __attribute__((ext_vector_type(8))) float v8f;

#define B_      4
#define N_      16384
#define BN_     (B_ * N_)
#define K_IMG   1024
#define EPS_BN  1e-5f

// ---------------------------------------------------------------------------
// Kernel 0: per-(b,c) image-row reductions.
//   g[b,c]    = mean_k img[b,c,k]
//   t[b,c]    = (1/K) sum_k b3[k] * img[b,c,k]
//   M[b,j,c]  = (1/K) sum_k W3[k,j] * img[b,c,k]     (64x64 per batch)
// Also zeroes the BN stat accumulators (384 floats).
// grid = 256 (= B*64), block = 64
// ---------------------------------------------------------------------------
__global__ __launch_bounds__(64) void prep_kernel(
    const float* __restrict__ img, const float* __restrict__ W3,
    const float* __restrict__ b3,
    float* __restrict__ g_ws, float* __restrict__ t_ws,
    float* __restrict__ M_ws, float* __restrict__ stats) {
  __shared__ float img_s[K_IMG];
  __shared__ float pg[64], pt[64];
  int b = blockIdx.x >> 6;
  int c = blockIdx.x & 63;
  int tid = threadIdx.x;

  const float* row = img + (size_t)(b * 64 + c) * K_IMG;
  for (int i = tid; i < K_IMG; i += 64) img_s[i] = row[i];
  __syncthreads();

  // M[b][j=tid][c]
  float accM = 0.0f;
  for (int k = 0; k < K_IMG; ++k) accM += W3[k * 64 + tid] * img_s[k];
  M_ws[b * 4096 + tid * 64 + c] = accM * (1.0f / K_IMG);

  // partial reductions for g and t
  float sg = 0.0f, st = 0.0f;
  int k0 = tid * 16;
  for (int k = k0; k < k0 + 16; ++k) { sg += img_s[k]; st += b3[k] * img_s[k]; }
  pg[tid] = sg; pt[tid] = st;
  __syncthreads();
  if (tid == 0) {
    float a = 0.0f, tt = 0.0f;
    for (int i = 0; i < 64; ++i) { a += pg[i]; tt += pt[i]; }
    g_ws[b * 64 + c] = a * (1.0f / K_IMG);
    t_ws[b * 64 + c] = tt * (1.0f / K_IMG);
  }
  if (blockIdx.x == 0) {
    for (int i = tid; i < 384; i += 64) stats[i] = 0.0f;
  }
}

// ---------------------------------------------------------------------------
// Kernel 1: y1[bn,128] = lidar[bn,64] @ W1[:,64:128]^T + c1[b,:]
//   c1[b,o] = b1[o] + W1[o,0:64] . g[b,:]   (computed per block, cheap)
// Accumulate per-channel sum and sum-of-squares for BN1 stats.
// grid = BN/64 = 1024, block = 256 (8 waves). Block tile: 64 rows x 128 cols.
// Wave w owns colTile w (16 cols), loops 4 row tiles.
// ---------------------------------------------------------------------------
__global__ __launch_bounds__(256) void mlp1_kernel(
    const float* __restrict__ lidar, const float* __restrict__ W1,
    const float* __restrict__ b1, const float* __restrict__ g_ws,
    float* __restrict__ y1, float* __restrict__ s1sum, float* __restrict__ s1sq) {
  __shared__ float As[64 * 66];      // lidar tile, padded
  __shared__ float Ws[128 * 66];     // W1[:,64:128], padded: Ws[o*66+c]
  __shared__ float g_s[64];
  __shared__ float c1_s[128];
  __shared__ float ssum[128], ssq[128];

  int tid = threadIdx.x;
  int R = blockIdx.x * 64;           // global row base (point index within BN)
  int b = R / N_;                    // N_ % 64 == 0 -> uniform per block

  const float* Ag = lidar + (size_t)R * 64;
  for (int i0 = tid * 4; i0 < 4096; i0 += 1024) {
    const float4 v = *(const float4*)(Ag + i0);
    float* dst = As + (i0 >> 6) * 66 + (i0 & 63);
    dst[0] = v.x; dst[1] = v.y; dst[2] = v.z; dst[3] = v.w;
  }
  for (int i0 = tid * 4; i0 < 8192; i0 += 1024) {
    int o = i0 >> 6, cc = i0 & 63;
    const float4 v = *(const float4*)(W1 + o * 128 + 64 + cc);
    float* dst = Ws + o * 66 + cc;
    dst[0] = v.x; dst[1] = v.y; dst[2] = v.z; dst[3] = v.w;
  }
  if (tid < 64) g_s[tid] = g_ws[b * 64 + tid];
  if (tid < 128) { ssum[tid] = 0.0f; ssq[tid] = 0.0f; }
  __syncthreads();
  if (tid < 128) {
    float a = b1[tid];
    const float* wrow = W1 + tid * 128;
    for (int c = 0; c < 64; ++c) a += wrow[c] * g_s[c];
    c1_s[tid] = a;
  }
  __syncthreads();

  int wave = tid >> 5, lane = tid & 31;
  int n = lane & 15;
  int koff = (lane >> 4) * 2;        // K offset for this half-wave
  int colBase = wave * 16;
  const float* brow = Ws + (colBase + n) * 66;
  float cadd = c1_s[colBase + n];

  for (int rt = 0; rt < 4; ++rt) {
    const float* arow = As + (rt * 16 + n) * 66;
    v8f acc = {0.f, 0.f, 0.f, 0.f, 0.f, 0.f, 0.f, 0.f};
#pragma unroll
    for (int kk = 0; kk < 16; ++kk) {
      int k0 = kk * 4 + koff;
      v2f av, bv;
      av[0] = arow[k0]; av[1] = arow[k0 + 1];
      bv[0] = brow[k0]; bv[1] = brow[k0 + 1];
      acc = __builtin_amdgcn_wmma_f32_16x16x4_f32(
          false, av, false, bv, (short)0, acc, false, false);
    }
    int rowBase = R + rt * 16 + 8 * (lane >> 4);
    float* orow = y1 + (size_t)rowBase * 128 + colBase + n;
    float s = 0.0f, q = 0.0f;
#pragma unroll
    for (int i = 0; i < 8; ++i) {
      float v = acc[i] + cadd;
      s += v; q += v * v;
      orow[(size_t)i * 128] = v;
    }
    atomicAdd(&ssum[colBase + n], s);
    atomicAdd(&ssq[colBase + n], q);
  }
  __syncthreads();
  if (tid < 128) {
    atomicAdd(&s1sum[tid], ssum[tid]);
    atomicAdd(&s1sq[tid], ssq[tid]);
  }
}

// ---------------------------------------------------------------------------
// Kernel 2: y2[bn,64] = relu(BN1(y1)) @ W2^T + b2 ; accumulate BN2 stats.
// grid = BN/32 = 2048, block = 256. Block tile: 32 rows x 64 cols.
// Wave w: rowTile = w>>2, colTile = w&3. K = 128.
// ---------------------------------------------------------------------------
__global__ __launch_bounds__(256) void mlp2_kernel(
    const float* __restrict__ y1, const float* __restrict__ W2,
    const float* __restrict__ b2, const float* __restrict__ g1v,
    const float* __restrict__ be1,
    const float* __restrict__ s1sum, const float* __restrict__ s1sq,
    float* __restrict__ y2, float* __restrict__ s2sum, float* __restrict__ s2sq) {
  __shared__ float As[32 * 130];
  __shared__ float Ws[64 * 130];
  __shared__ float sc1[128], sh1[128];
  __shared__ float ssum[64], ssq[64];

  int tid = threadIdx.x;
  int R = blockIdx.x * 32;

  if (tid < 128) {
    float mu  = s1sum[tid] * (1.0f / BN_);
    float var = s1sq[tid] * (1.0f / BN_) - mu * mu;
    float sc  = g1v[tid] * rsqrtf(var + EPS_BN);
    sc1[tid] = sc;
    sh1[tid] = be1[tid] - mu * sc;
  }
  if (tid < 64) { ssum[tid] = 0.0f; ssq[tid] = 0.0f; }
  for (int i0 = tid * 4; i0 < 8192; i0 += 1024) {
    int o = i0 >> 7, cc = i0 & 127;
    const float4 v = *(const float4*)(W2 + o * 128 + cc);
    float* dst = Ws + o * 130 + cc;
    dst[0] = v.x; dst[1] = v.y; dst[2] = v.z; dst[3] = v.w;
  }
  __syncthreads();
  const float* Ag = y1 + (size_t)R * 128;
  for (int i0 = tid * 4; i0 < 4096; i0 += 1024) {
    const float4 v = *(const float4*)(Ag + i0);
    int r = i0 >> 7, cc = i0 & 127;
    float* dst = As + r * 130 + cc;
    dst[0] = fmaxf(v.x * sc1[cc]     + sh1[cc],     0.0f);
    dst[1] = fmaxf(v.y * sc1[cc + 1] + sh1[cc + 1], 0.0f);
    dst[2] = fmaxf(v.z * sc1[cc + 2] + sh1[cc + 2], 0.0f);
    dst[3] = fmaxf(v.w * sc1[cc + 3] + sh1[cc + 3], 0.0f);
  }
  __syncthreads();

  int wave = tid >> 5, lane = tid & 31;
  int n = lane & 15, koff = (lane >> 4) * 2;
  int rt = wave >> 2;
  int colBase = (wave & 3) * 16;
  const float* brow = Ws + (colBase + n) * 130;
  const float* arow = As + (rt * 16 + n) * 130;
  float badd = b2[colBase + n];

  v8f acc = {0.f, 0.f, 0.f, 0.f, 0.f, 0.f, 0.f, 0.f};
#pragma unroll
  for (int kk = 0; kk < 32; ++kk) {
    int k0 = kk * 4 + koff;
    v2f av, bv;
    av[0] = arow[k0]; av[1] = arow[k0 + 1];
    bv[0] = brow[k0]; bv[1] = brow[k0 + 1];
    acc = __builtin_amdgcn_wmma_f32_16x16x4_f32(
        false, av, false, bv, (short)0, acc, false, false);
  }
  int rowBase = R + rt * 16 + 8 * (lane >> 4);
  float* orow = y2 + (size_t)rowBase * 64 + colBase + n;
  float s = 0.0f, q = 0.0f;
#pragma unroll
  for (int i = 0; i < 8; ++i) {
    float v = acc[i] + badd;
    s += v; q += v * v;
    orow[(size_t)i * 64] = v;
  }
  atomicAdd(&ssum[colBase + n], s);
  atomicAdd(&ssq[colBase + n], q);
  __syncthreads();
  if (tid < 64) {
    atomicAdd(&s2sum[tid], ssum[tid]);
    atomicAdd(&s2sq[tid], ssq[tid]);
  }
}

// ---------------------------------------------------------------------------
// Kernel 3: out[bn,64] = relu(BN2(y2)) @ M[b] + t[b]
// grid = BN/64 = 1024, block = 256. Block tile: 64 rows x 64 cols. K = 64.
// Wave w: colTile = w&3, rowTiles = 2*(w>>2)+{0,1}.
// ---------------------------------------------------------------------------
__global__ __launch_bounds__(256) void mlp3_kernel(
    const float* __restrict__ y2, const float* __restrict__ M_ws,
    const float* __restrict__ t_ws, const float* __restrict__ g2v,
    const float* __restrict__ be2,
    const float* __restrict__ s2sum, const float* __restrict__ s2sq,
    float* __restrict__ out) {
  __shared__ float As[64 * 66];
  __shared__ float Ms[64 * 66];      // Ms[j*66 + c] = M[b][j][c]
  __shared__ float sc2[64], sh2[64], t_s[64];

  int tid = threadIdx.x;
  int R = blockIdx.x * 64;
  int b = R / N_;

  if (tid < 64) {
    float mu  = s2sum[tid] * (1.0f / BN_);
    float var = s2sq[tid] * (1.0f / BN_) - mu * mu;
    float sc  = g2v[tid] * rsqrtf(var + EPS_BN);
    sc2[tid] = sc;
    sh2[tid] = be2[tid] - mu * sc;
    t_s[tid] = t_ws[b * 64 + tid];
  }
  const float* Mg = M_ws + b * 4096;
  for (int i0 = tid * 4; i0 < 4096; i0 += 1024) {
    const float4 v = *(const float4*)(Mg + i0);
    float* dst = Ms + (i0 >> 6) * 66 + (i0 & 63);
    dst[0] = v.x; dst[1] = v.y; dst[2] = v.z; dst[3] = v.w;
  }
  __syncthreads();
  const float* Ag = y2 + (size_t)R * 64;
  for (int i0 = tid * 4; i0 < 4096; i0 += 1024) {
    const float4 v = *(const float4*)(Ag + i0);
    int r = i0 >> 6, cc = i0 & 63;
    float* dst = As + r * 66 + cc;
    dst[0] = fmaxf(v.x * sc2[cc]     + sh2[cc],     0.0f);
    dst[1] = fmaxf(v.y * sc2[cc + 1] + sh2[cc + 1], 0.0f);
    dst[2] = fmaxf(v.z * sc2[cc + 2] + sh2[cc + 2], 0.0f);
    dst[3] = fmaxf(v.w * sc2[cc + 3] + sh2[cc + 3], 0.0f);
  }
  __syncthreads();

  int wave = tid >> 5, lane = tid & 31;
  int n = lane & 15, koff = (lane >> 4) * 2;
  int colBase = (wave & 3) * 16;
  const float* bcol = Ms + colBase + n;   // B(k,n) = Ms[k*66 + colBase+n]
  float tadd = t_s[colBase + n];

  for (int rti = 0; rti < 2; ++rti) {
    int rt = (wave >> 2) * 2 + rti;
    const float* arow = As + (rt * 16 + n) * 66;
    v8f acc = {0.f, 0.f, 0.f, 0.f, 0.f, 0.f, 0.f, 0.f};
#pragma unroll
    for (int kk = 0; kk < 16; ++kk) {
      int k0 = kk * 4 + koff;
      v2f av, bv;
      av[0] = arow[k0];       av[1] = arow[k0 + 1];
      bv[0] = bcol[k0 * 66];  bv[1] = bcol[(k0 + 1) * 66];
      acc = __builtin_amdgcn_wmma_f32_16x16x4_f32(
          false, av, false, bv, (short)0, acc, false, false);
    }
    int rowBase = R + rt * 16 + 8 * (lane >> 4);
    float* orow = out + (size_t)rowBase * 64 + colBase + n;
#pragma unroll
    for (int i = 0; i < 8; ++i) orow[(size_t)i * 64] = acc[i] + tadd;
  }
}

// ---------------------------------------------------------------------------
extern "C" void kernel_launch(void* const* d_in, const int* in_sizes, int n_in,
                              void* d_out, int out_size, void* d_ws, size_t ws_size,
                              hipStream_t stream) {
  const float* lidar = (const float*)d_in[0];
  const float* img   = (const float*)d_in[1];
  const float* W1    = (const float*)d_in[2];
  const float* b1    = (const float*)d_in[3];
  const float* g1    = (const float*)d_in[4];
  const float* be1   = (const float*)d_in[5];
  const float* W2    = (const float*)d_in[6];
  const float* b2    = (const float*)d_in[7];
  const float* g2    = (const float*)d_in[8];
  const float* be2   = (const float*)d_in[9];
  const float* W3    = (const float*)d_in[10];
  const float* b3    = (const float*)d_in[11];

  float* ws    = (float*)d_ws;
  float* y1    = ws;                             // BN_*128 floats (32 MB)
  float* y2    = y1 + (size_t)BN_ * 128;         // BN_*64 floats (16 MB)
  float* g_ws  = y2 + (size_t)BN_ * 64;          // 256
  float* t_ws  = g_ws + 256;                     // 256
  float* M_ws  = t_ws + 256;                     // B*64*64 = 16384
  float* stats = M_ws + B_ * 4096;               // 384: s1sum|s1sq|s2sum|s2sq
  float* s1sum = stats;
  float* s1sq  = stats + 128;
  float* s2sum = stats + 256;
  float* s2sq  = stats + 320;

  prep_kernel<<<256, 64, 0, stream>>>(img, W3, b3, g_ws, t_ws, M_ws, stats);
  mlp1_kernel<<<BN_ / 64, 256, 0, stream>>>(lidar, W1, b1, g_ws, y1, s1sum, s1sq);
  mlp2_kernel<<<BN_ / 32, 256, 0, stream>>>(y1, W2, b2, g1, be1, s1sum, s1sq,
                                            y2, s2sum, s2sq);
  mlp3_kernel<<<BN_ / 64, 256, 0, stream>>>(y2, M_ws, t_ws, g2, be2,
                                            s2sum, s2sq, (float*)d_out);
}